// FusedSNNSSMLayer_23888608100423
// MI455X (gfx1250) — compile-verified
//
#include <hip/hip_runtime.h>

#define V_DECAY 0.9f
#define V_THRESH 1.0f
#define REFRAC 5.0f
#define B_SZ 8
#define SEQ 4096
#define HID 1024
#define STATE 64

typedef __attribute__((ext_vector_type(2))) float v2f;
typedef __attribute__((ext_vector_type(8))) float v8f;

// ---------------------------------------------------------------------------
// Kernel 1: LIF neuron scan, pure elementwise recurrence (no reductions, no
// barriers). 8192 independent (b,h) chains, one per thread, coalesced streams.
// Spikes go straight to d_out. global_prefetch_b8 issued 16 steps ahead.
// ---------------------------------------------------------------------------
__global__ void __launch_bounds__(256) snn_scan(const float* __restrict__ x,
                                                float* __restrict__ out) {
    const int gid = blockIdx.x * 256 + threadIdx.x;   // 0..8191
    const int b   = gid >> 10;
    const int h   = gid & (HID - 1);

    float v = 0.0f, r = 0.0f;
    const float* xb = x   + (size_t)b * SEQ * HID + h;
    float*       ob = out + (size_t)b * SEQ * HID + h;

    #pragma unroll 4
    for (int t = 0; t < SEQ; ++t) {
        int tp = t + 16; tp = (tp < SEQ) ? tp : (SEQ - 1);
        __builtin_prefetch(xb + (size_t)tp * HID, 0, 1);   // global_prefetch_b8
        float xt = xb[(size_t)t * HID];
        v = v * V_DECAY + xt;
        float spike = ((r <= 0.0f) && (v >= V_THRESH)) ? 1.0f : 0.0f;
        v = (spike > 0.0f) ? 0.0f : v;
        r = (spike > 0.0f) ? REFRAC : fmaxf(r - 1.0f, 0.0f);
        ob[(size_t)t * HID] = spike;
    }
}

// ---------------------------------------------------------------------------
// Kernel 2: u[b,t] = mean_h spike. One wave per (b,t) row of 1024 floats,
// float4 fully-coalesced loads + wave32 shuffle reduce. Bandwidth-bound.
// ---------------------------------------------------------------------------
__global__ void __launch_bounds__(256) spike_mean(const float* __restrict__ out,
                                                  float* __restrict__ u) {
    const int wid  = (blockIdx.x * 256 + threadIdx.x) >> 5;  // 0..B_SZ*SEQ-1
    const int lane = threadIdx.x & 31;
    const float* row = out + (size_t)wid * HID;
    float s = 0.0f;
    #pragma unroll
    for (int j = 0; j < 8; ++j) {
        float4 vv = *(const float4*)(row + 4 * (lane + 32 * j));
        s += vv.x + vv.y + vv.z + vv.w;
    }
    #pragma unroll
    for (int off = 16; off > 0; off >>= 1) s += __shfl_xor(s, off, 32);
    if (lane == 0) u[wid] = s * (1.0f / (float)HID);
}

// ---------------------------------------------------------------------------
// Kernel 3: impulse response g[j] = C A^j B (+D at j=0), log-depth:
//   A^64 by 6 in-place LDS squarings; P_s = A^s B (64-step chain);
//   w_q = C A^{64q} (64-step chain) fused with G[q][s] = w_q . P_s.
// One block, 1024 threads, ~49KB LDS, conflict-free access patterns.
// ---------------------------------------------------------------------------
__global__ void __launch_bounds__(1024) ssm_impulse(const float* __restrict__ A,
                                                    const float* __restrict__ Bv,
                                                    const float* __restrict__ Cv,
                                                    const float* __restrict__ Dv,
                                                    float* __restrict__ gpad) {
    __shared__ float sM [STATE * STATE];  // A, then A^2 ... A^64 (row major)
    __shared__ float sAT[STATE * STATE];  // sAT[k*64+n] = A[n][k]
    __shared__ float sPT[STATE * STATE];  // sPT[n*64+s] = (A^s B)[n]
    __shared__ float w  [STATE];          // current w_q = C A^{64q}

    const int tid = threadIdx.x;
    const float Dsc = Dv[0];

    for (int i = tid; i < STATE * STATE; i += 1024) sM[i] = A[i];
    __syncthreads();
    for (int i = tid; i < STATE * STATE; i += 1024) {
        int k = i >> 6, n = i & 63;
        sAT[i] = sM[n * STATE + k];
    }
    __syncthreads();

    // ---- sM <- A^64 via 6 in-place squarings (register staged) ----
    const int sc = tid & 63;        // output col
    const int n0 = tid >> 6;        // 0..15 -> rows n0+16*i
    for (int it = 0; it < 6; ++it) {
        float acc[4] = {0.f, 0.f, 0.f, 0.f};
        for (int k = 0; k < STATE; ++k) {
            float mks = sM[k * STATE + sc];            // consecutive lanes
            #pragma unroll
            for (int i = 0; i < 4; ++i)                // M[n][k]: broadcast
                acc[i] = fmaf(sM[(n0 + 16 * i) * STATE + k], mks, acc[i]);
        }
        __syncthreads();                               // all reads done
        #pragma unroll
        for (int i = 0; i < 4; ++i) sM[(n0 + 16 * i) * STATE + sc] = acc[i];
        __syncthreads();
    }

    // ---- init P_0 = B, w_0 = C, gpad front pad ----
    if (tid < STATE)  sPT[tid * STATE + 0] = Bv[tid];
    else if (tid < 2 * STATE) w[tid - STATE] = Cv[tid - STATE];
    if (tid < 16) gpad[tid] = 0.0f;
    __syncthreads();

    // ---- P chain: P_s = A * P_{s-1}  (writes col s, reads col s-1) ----
    for (int s = 1; s < STATE; ++s) {
        if (tid < STATE) {
            float acc = 0.0f;
            for (int k = 0; k < STATE; ++k)            // sAT consec, sPT bcast
                acc = fmaf(sAT[k * STATE + tid], sPT[k * STATE + (s - 1)], acc);
            sPT[tid * STATE + s] = acc;
        }
        __syncthreads();
    }

    // ---- q loop: emit G row q, advance w ----
    for (int q = 0; q < STATE; ++q) {
        float accG = 0.0f, accW = 0.0f;
        if (tid < STATE) {                             // G[q][s], s = tid
            for (int n = 0; n < STATE; ++n)            // w bcast, sPT consec
                accG = fmaf(w[n], sPT[n * STATE + tid], accG);
            int j = q * STATE + tid;
            gpad[16 + j] = accG + ((j == 0) ? Dsc : 0.0f);
        } else if (tid < 2 * STATE) {                  // w_{q+1}[n], n = tid-64
            const int n = tid - STATE;
            for (int k = 0; k < STATE; ++k)            // w bcast, sM consec
                accW = fmaf(w[k], sM[k * STATE + n], accW);
        }
        __syncthreads();
        if (tid >= STATE && tid < 2 * STATE) w[tid - STATE] = accW;
        __syncthreads();
    }
}

// ---------------------------------------------------------------------------
// Kernel 4: causal Toeplitz convolution y[b,t] = sum_j g[j] u[b,t-j] via
// V_WMMA_F32_16X16X4_F32. One wave per 16-timestep tile (EXEC all ones).
// ---------------------------------------------------------------------------
__global__ void __launch_bounds__(32) conv_wmma(const float* __restrict__ gpad,
                                                const float* __restrict__ u,
                                                float* __restrict__ y) {
    const int t0    = blockIdx.x * 16;
    const int l     = threadIdx.x;
    const int half  = l >> 4;
    const int m     = l & 15;              // A row (t) / B col (batch)
    const int kbase = half * 2;

    v8f acc = {};
    const int nchunks = t0 / 4 + 4;        // cover s = 0 .. t0+15

    const float* gA = gpad + 16 + t0 + m - kbase;
    for (int c = 0; c < nchunks; ++c) {
        const int s0 = 4 * c;
        v2f a;                             // Toeplitz slice of g
        a.x = gA[-s0];
        a.y = gA[-s0 - 1];
        v2f bf;                            // u tile, batches >= 8 zero
        bf.x = (m < B_SZ) ? u[m * SEQ + s0 + kbase]     : 0.0f;
        bf.y = (m < B_SZ) ? u[m * SEQ + s0 + kbase + 1] : 0.0f;
        acc = __builtin_amdgcn_wmma_f32_16x16x4_f32(
            false, a, false, bf, (short)0, acc, false, false);
    }

    if (m < B_SZ) {
        const float* af = (const float*)&acc;
        #pragma unroll
        for (int v = 0; v < 8; ++v)
            y[m * SEQ + t0 + v + half * 8] = af[v];
    }
}

// ---------------------------------------------------------------------------
// Kernel 5: out[b,t,h] = spike (already stored) + y[b,t]
// ---------------------------------------------------------------------------
__global__ void add_y(float* __restrict__ out, const float* __restrict__ y) {
    size_t i = (size_t)blockIdx.x * blockDim.x + threadIdx.x;
    int bt = (int)(i >> 10);               // b*SEQ + t  (HID = 1024)
    out[i] += y[bt];
}

extern "C" void kernel_launch(void* const* d_in, const int* in_sizes, int n_in,
                              void* d_out, int out_size, void* d_ws, size_t ws_size,
                              hipStream_t stream) {
    (void)in_sizes; (void)n_in; (void)out_size; (void)ws_size;
    const float* x  = (const float*)d_in[0];   // (8, 4096, 1024)
    const float* A  = (const float*)d_in[1];   // (64, 64)
    const float* Bv = (const float*)d_in[2];   // (64, 1)
    const float* Cv = (const float*)d_in[3];   // (1, 64)
    const float* Dv = (const float*)d_in[4];   // (1, 1)
    float* out = (float*)d_out;
    float* ws  = (float*)d_ws;

    float* u    = ws;                          // 8*4096 floats
    float* gpad = ws + B_SZ * SEQ;             // 16 + 4096 floats
    float* y    = ws + B_SZ * SEQ + 4128;      // 8*4096 floats (~272 KB total)

    snn_scan   <<<(B_SZ * HID) / 256, 256, 0, stream>>>(x, out);
    spike_mean <<<(B_SZ * SEQ * 32) / 256, 256, 0, stream>>>(out, u);
    ssm_impulse<<<1, 1024, 0, stream>>>(A, Bv, Cv, Dv, gpad);
    conv_wmma  <<<SEQ / 16, 32, 0, stream>>>(gpad, u, y);
    add_y      <<<(B_SZ * SEQ * HID) / 256, 256, 0, stream>>>(out, y);
}